// MyModel_71227737637138
// MI455X (gfx1250) — compile-verified
//
#include <hip/hip_runtime.h>
#include <hip/hip_bf16.h>

// Problem constants (match reference).
#define NN 100000
#define EE 1000000
#define DD 64
#define D2 128
#define LLAY 3
#define BB 2048
#define KK 4096

typedef float v2f __attribute__((ext_vector_type(2)));
typedef float v8f __attribute__((ext_vector_type(8)));

__device__ __forceinline__ v8f v8f_zero() {
    v8f z;
#pragma unroll
    for (int i = 0; i < 8; i++) z[i] = 0.0f;
    return z;
}

// D = A(16x4 f32) * B(4x16 f32) + C(16x16 f32). Native f32 WMMA on CDNA5.
__device__ __forceinline__ v8f wmma_f32(v2f a, v2f b, v8f c) {
    return __builtin_amdgcn_wmma_f32_16x16x4_f32(false, a, false, b, (short)0, c, false, false);
}

// ---------------------------------------------------------------------------
// 1) Edge scatter-add: agg[dst] += h[src], one thread per (edge, dim).
// ---------------------------------------------------------------------------
__global__ void k_scatter_add(const float* __restrict__ h,
                              const int* __restrict__ src,
                              const int* __restrict__ dst,
                              float* __restrict__ agg, int nE) {
    long long t = (long long)blockIdx.x * blockDim.x + threadIdx.x;
    long long total = (long long)nE * DD;
    if (t >= total) return;
    int e = (int)(t >> 6);
    int d = (int)(t & 63);
    atomicAdd(&agg[(long long)dst[e] * DD + d], h[(long long)src[e] * DD + d]);
}

// ---------------------------------------------------------------------------
// 2) Fused GIN layer (one wave handles 16 rows):
//    z = (1+eps)*h + agg ; hid = relu(z@W1+b1) ; o = hid@W2+b2 ; h' = o*g+bb [relu]
// ---------------------------------------------------------------------------
__global__ void __launch_bounds__(32)
k_gin_layer(const float* __restrict__ hin, const float* __restrict__ agg,
            const float* __restrict__ W1, const float* __restrict__ b1,
            const float* __restrict__ W2, const float* __restrict__ b2,
            const float* __restrict__ g, const float* __restrict__ bb,
            const float* __restrict__ epsArr, int layer, int do_relu,
            float* __restrict__ hout, int n) {
    __shared__ float zsm[16 * DD];   // 4 KB
    __shared__ float hid[16 * D2];   // 8 KB
    const int lane = threadIdx.x;
    const int row0 = blockIdx.x * 16;
    const float eps1 = 1.0f + epsArr[layer];
    const float* W1l = W1 + (size_t)layer * DD * D2;
    const float* b1l = b1 + (size_t)layer * D2;
    const float* W2l = W2 + (size_t)layer * D2 * DD;
    const float* b2l = b2 + (size_t)layer * DD;
    const float* gl  = g  + (size_t)layer * DD;
    const float* bbl = bb + (size_t)layer * DD;

    // Cooperative, guarded load of the 16x64 z tile into LDS.
    for (int i = lane; i < 16 * 16; i += 32) {
        int r = i >> 4, c4 = (i & 15) * 4;
        int row = row0 + r;
        float4 z = make_float4(0.f, 0.f, 0.f, 0.f);
        if (row < n) {
            float4 hv = *(const float4*)(hin + (size_t)row * DD + c4);
            float4 av = *(const float4*)(agg + (size_t)row * DD + c4);
            z = make_float4(eps1 * hv.x + av.x, eps1 * hv.y + av.y,
                            eps1 * hv.z + av.z, eps1 * hv.w + av.w);
        }
        *(float4*)&zsm[r * DD + c4] = z;
    }
    __syncthreads();

    const int half = lane >> 4;   // K-pair half (k0+2*half, k0+2*half+1) per f32 WMMA layout
    const int l16  = lane & 15;

    // GEMM1: 16x64 @ 64x128 -> 8 accumulators of 16x16.
    v8f c[8];
#pragma unroll
    for (int t = 0; t < 8; t++) c[t] = v8f_zero();
#pragma unroll
    for (int k0 = 0; k0 < DD; k0 += 4) {
        v2f a;
        a.x = zsm[l16 * DD + k0 + 2 * half + 0];
        a.y = zsm[l16 * DD + k0 + 2 * half + 1];
#pragma unroll
        for (int t = 0; t < 8; t++) {
            int nb = t * 16 + l16;
            v2f b;
            b.x = W1l[(k0 + 2 * half + 0) * D2 + nb];
            b.y = W1l[(k0 + 2 * half + 1) * D2 + nb];
            c[t] = wmma_f32(a, b, c[t]);
        }
    }
#pragma unroll
    for (int t = 0; t < 8; t++) {
        int nb = t * 16 + l16;
        float bias = b1l[nb];
#pragma unroll
        for (int v = 0; v < 8; v++)
            hid[(v + 8 * half) * D2 + nb] = fmaxf(c[t][v] + bias, 0.0f);
    }
    __syncthreads();

    // GEMM2: 16x128 @ 128x64 -> 4 accumulators.
    v8f d[4];
#pragma unroll
    for (int t = 0; t < 4; t++) d[t] = v8f_zero();
#pragma unroll
    for (int k0 = 0; k0 < D2; k0 += 4) {
        v2f a;
        a.x = hid[l16 * D2 + k0 + 2 * half + 0];
        a.y = hid[l16 * D2 + k0 + 2 * half + 1];
#pragma unroll
        for (int t = 0; t < 4; t++) {
            int nb = t * 16 + l16;
            v2f b;
            b.x = W2l[(k0 + 2 * half + 0) * DD + nb];
            b.y = W2l[(k0 + 2 * half + 1) * DD + nb];
            d[t] = wmma_f32(a, b, d[t]);
        }
    }
#pragma unroll
    for (int t = 0; t < 4; t++) {
        int nb = t * 16 + l16;
        float b2v = b2l[nb], gv = gl[nb], bbv = bbl[nb];
#pragma unroll
        for (int v = 0; v < 8; v++) {
            int row = row0 + v + 8 * half;
            if (row < n) {
                float val = (d[t][v] + b2v) * gv + bbv;
                if (do_relu) val = fmaxf(val, 0.0f);
                hout[(size_t)row * DD + nb] = val;
            }
        }
    }
}

// ---------------------------------------------------------------------------
// 3) Separator head: score = sigmoid( relu((nf@sW1+sb1)*sg+sb) @ sW2 + sb2 )
// ---------------------------------------------------------------------------
__global__ void __launch_bounds__(32)
k_sep_head(const float* __restrict__ nf,
           const float* __restrict__ W1, const float* __restrict__ b1,
           const float* __restrict__ sg, const float* __restrict__ sb,
           const float* __restrict__ W2, const float* __restrict__ b2,
           float* __restrict__ score, int n) {
    __shared__ float zsm[16 * DD];
    __shared__ float hid[16 * D2];
    const int lane = threadIdx.x;
    const int row0 = blockIdx.x * 16;

    for (int i = lane; i < 16 * 16; i += 32) {
        int r = i >> 4, c4 = (i & 15) * 4;
        int row = row0 + r;
        float4 z = make_float4(0.f, 0.f, 0.f, 0.f);
        if (row < n) z = *(const float4*)(nf + (size_t)row * DD + c4);
        *(float4*)&zsm[r * DD + c4] = z;
    }
    __syncthreads();

    const int half = lane >> 4, l16 = lane & 15;

    v8f c[8];
#pragma unroll
    for (int t = 0; t < 8; t++) c[t] = v8f_zero();
#pragma unroll
    for (int k0 = 0; k0 < DD; k0 += 4) {
        v2f a;
        a.x = zsm[l16 * DD + k0 + 2 * half + 0];
        a.y = zsm[l16 * DD + k0 + 2 * half + 1];
#pragma unroll
        for (int t = 0; t < 8; t++) {
            int nb = t * 16 + l16;
            v2f b;
            b.x = W1[(k0 + 2 * half + 0) * D2 + nb];
            b.y = W1[(k0 + 2 * half + 1) * D2 + nb];
            c[t] = wmma_f32(a, b, c[t]);
        }
    }
#pragma unroll
    for (int t = 0; t < 8; t++) {
        int nb = t * 16 + l16;
        float bias = b1[nb], gv = sg[nb], bv = sb[nb];
#pragma unroll
        for (int v = 0; v < 8; v++)
            hid[(v + 8 * half) * D2 + nb] = fmaxf((c[t][v] + bias) * gv + bv, 0.0f);
    }
    __syncthreads();

    v8f d[4];
#pragma unroll
    for (int t = 0; t < 4; t++) d[t] = v8f_zero();
#pragma unroll
    for (int k0 = 0; k0 < D2; k0 += 4) {
        v2f a;
        a.x = hid[l16 * D2 + k0 + 2 * half + 0];
        a.y = hid[l16 * D2 + k0 + 2 * half + 1];
#pragma unroll
        for (int t = 0; t < 4; t++) {
            int nb = t * 16 + l16;
            v2f b;
            b.x = W2[(k0 + 2 * half + 0) * DD + nb];
            b.y = W2[(k0 + 2 * half + 1) * DD + nb];
            d[t] = wmma_f32(a, b, d[t]);
        }
    }
#pragma unroll
    for (int t = 0; t < 4; t++) {
        int nb = t * 16 + l16;
        float b2v = b2[nb];
#pragma unroll
        for (int v = 0; v < 8; v++) {
            int row = row0 + v + 8 * half;
            if (row < n) {
                float val = d[t][v] + b2v;
                score[(size_t)row * DD + nb] = 1.0f / (1.0f + __expf(-val));
            }
        }
    }
}

// ---------------------------------------------------------------------------
// 4) Codebook squared norms.
// ---------------------------------------------------------------------------
__global__ void k_cb_norm(const float* __restrict__ cb, float* __restrict__ cbn) {
    int k = blockIdx.x * blockDim.x + threadIdx.x;
    if (k >= KK) return;
    const float4* r = (const float4*)(cb + (size_t)k * DD);
    float s = 0.f;
#pragma unroll
    for (int j = 0; j < 16; j++) {
        float4 v = r[j];
        s += v.x * v.x + v.y * v.y + v.z * v.z + v.w * v.w;
    }
    cbn[k] = s;
}

// ---------------------------------------------------------------------------
// 5) VQ nearest-code search: the 52-GFLOP GEMM nf @ codebook^T via WMMA.
//    Track min over (||c||^2 - 2*nf.c); ||nf||^2 added in finalize.
// ---------------------------------------------------------------------------
__global__ void __launch_bounds__(32)
k_vq(const float* __restrict__ nf, const float* __restrict__ cb,
     const float* __restrict__ cbn, int* __restrict__ vqidx,
     float* __restrict__ minT, int n) {
    __shared__ float zsm[16 * DD];
    const int lane = threadIdx.x;
    const int row0 = blockIdx.x * 16;

    for (int i = lane; i < 16 * 16; i += 32) {
        int r = i >> 4, c4 = (i & 15) * 4;
        int row = row0 + r;
        float4 z = make_float4(0.f, 0.f, 0.f, 0.f);
        if (row < n) z = *(const float4*)(nf + (size_t)row * DD + c4);
        *(float4*)&zsm[r * DD + c4] = z;
    }
    __syncthreads();

    const int half = lane >> 4, l16 = lane & 15;

    float minv[8];
    int mini[8];
#pragma unroll
    for (int v = 0; v < 8; v++) { minv[v] = 3.4e38f; mini[v] = 0; }

    for (int c0 = 0; c0 < KK; c0 += 16) {
        const float* cbrow = cb + (size_t)(c0 + l16) * DD;
        __builtin_prefetch(cbrow + 16 * DD, 0, 1);  // next codebook tile
        v8f acc = v8f_zero();
#pragma unroll
        for (int k0 = 0; k0 < DD; k0 += 4) {
            v2f a;
            a.x = zsm[l16 * DD + k0 + 2 * half + 0];
            a.y = zsm[l16 * DD + k0 + 2 * half + 1];
            v2f b;   // B[k][j] = codebook[c0+j][k]
            b.x = cbrow[k0 + 2 * half + 0];
            b.y = cbrow[k0 + 2 * half + 1];
            acc = wmma_f32(a, b, acc);
        }
        float cnv = cbn[c0 + l16];
#pragma unroll
        for (int v = 0; v < 8; v++) {
            float t = cnv - 2.0f * acc[v];
            if (t < minv[v]) { minv[v] = t; mini[v] = c0 + l16; }
        }
    }
    // Reduce (min, idx) across the 16 lanes of each half.
#pragma unroll
    for (int off = 1; off < 16; off <<= 1) {
#pragma unroll
        for (int v = 0; v < 8; v++) {
            float ov = __shfl_xor(minv[v], off, 32);
            int oi = __shfl_xor(mini[v], off, 32);
            if (ov < minv[v]) { minv[v] = ov; mini[v] = oi; }
        }
    }
    if (l16 == 0) {
#pragma unroll
        for (int v = 0; v < 8; v++) {
            int row = row0 + v + 8 * half;
            if (row < n) { vqidx[row] = mini[v]; minT[row] = minv[v]; }
        }
    }
}

// ---------------------------------------------------------------------------
// 6) Per-node score reduction and per-graph counters.
// ---------------------------------------------------------------------------
__global__ void k_score_reduce(const float* __restrict__ score,
                               const int* __restrict__ batch,
                               float* __restrict__ posb, float* __restrict__ negb,
                               float* __restrict__ cnt, int n) {
    int i = blockIdx.x * blockDim.x + threadIdx.x;
    if (i >= n) return;
    const float4* sp = (const float4*)(score + (size_t)i * DD);
    float s = 0.f;
#pragma unroll
    for (int j = 0; j < 16; j++) {
        float4 v = sp[j];
        s += v.x + v.y + v.z + v.w;
    }
    s *= (1.0f / 64.0f);
    int b = batch[i];
    atomicAdd(&posb[b], s);
    atomicAdd(&negb[b], 1.0f - s);
    atomicAdd(&cnt[b], 1.0f);
}

// ---------------------------------------------------------------------------
// 7) Finalize nodes: res = nf + codebook[idx]; pool c/s; commitment accum.
// ---------------------------------------------------------------------------
__global__ void k_finalize(const float* __restrict__ nf, const float* __restrict__ cb,
                           const int* __restrict__ vqidx, const float* __restrict__ minT,
                           const float* __restrict__ score, const int* __restrict__ batch,
                           float* __restrict__ cpool, float* __restrict__ spool,
                           float* __restrict__ cmt_acc, int n) {
    int i = blockIdx.x * blockDim.x + threadIdx.x;
    if (i >= n) return;
    const float* nr = nf + (size_t)i * DD;
    const float* cr = cb + (size_t)vqidx[i] * DD;
    const float* sr = score + (size_t)i * DD;
    int b = batch[i];
    float* cp = cpool + (size_t)b * DD;
    float* sp = spool + (size_t)b * DD;
    float nfn = 0.f;
#pragma unroll 4
    for (int d = 0; d < DD; d++) {
        float nv = nr[d];
        nfn += nv * nv;
        float res = nv + cr[d];   // nf + vq_out(forward) == nf + codebook[idx]
        float sc = sr[d];
        atomicAdd(&cp[d], res * sc);
        atomicAdd(&sp[d], res * (1.0f - sc));
    }
    atomicAdd(cmt_acc, fmaxf(nfn + minT[i], 0.0f));  // ||nf - c||^2
}

// ---------------------------------------------------------------------------
// 8) Pool normalization into d_out.
// ---------------------------------------------------------------------------
__global__ void k_pool_norm(const float* __restrict__ cpool, const float* __restrict__ spool,
                            const float* __restrict__ cnt,
                            float* __restrict__ outc, float* __restrict__ outs, int total) {
    int t = blockIdx.x * blockDim.x + threadIdx.x;
    if (t >= total) return;
    float c = fmaxf(cnt[t >> 6], 1.0f);
    outc[t] = cpool[t] / c;
    outs[t] = spool[t] / c;
}

// ---------------------------------------------------------------------------
// 9) Classifier head on c_pool (WMMA GEMM1 + wave matvec for [128x1] GEMM2).
// ---------------------------------------------------------------------------
__global__ void __launch_bounds__(32)
k_classifier(const float* __restrict__ cp,
             const float* __restrict__ W1, const float* __restrict__ b1,
             const float* __restrict__ cg, const float* __restrict__ cbv,
             const float* __restrict__ W2, const float* __restrict__ b2,
             float* __restrict__ logit, int nb_rows) {
    __shared__ float zsm[16 * DD];
    __shared__ float hid[16 * D2];
    const int lane = threadIdx.x;
    const int row0 = blockIdx.x * 16;

    for (int i = lane; i < 16 * 16; i += 32) {
        int r = i >> 4, c4 = (i & 15) * 4;
        int row = row0 + r;
        float4 z = make_float4(0.f, 0.f, 0.f, 0.f);
        if (row < nb_rows) z = *(const float4*)(cp + (size_t)row * DD + c4);
        *(float4*)&zsm[r * DD + c4] = z;
    }
    __syncthreads();

    const int half = lane >> 4, l16 = lane & 15;

    v8f c[8];
#pragma unroll
    for (int t = 0; t < 8; t++) c[t] = v8f_zero();
#pragma unroll
    for (int k0 = 0; k0 < DD; k0 += 4) {
        v2f a;
        a.x = zsm[l16 * DD + k0 + 2 * half + 0];
        a.y = zsm[l16 * DD + k0 + 2 * half + 1];
#pragma unroll
        for (int t = 0; t < 8; t++) {
            int nb = t * 16 + l16;
            v2f b;
            b.x = W1[(k0 + 2 * half + 0) * D2 + nb];
            b.y = W1[(k0 + 2 * half + 1) * D2 + nb];
            c[t] = wmma_f32(a, b, c[t]);
        }
    }
#pragma unroll
    for (int t = 0; t < 8; t++) {
        int nb = t * 16 + l16;
        float bias = b1[nb], gv = cg[nb], bv = cbv[nb];
#pragma unroll
        for (int v = 0; v < 8; v++)
            hid[(v + 8 * half) * D2 + nb] = fmaxf((c[t][v] + bias) * gv + bv, 0.0f);
    }
    __syncthreads();

    // GEMM2 is [16x128] @ [128x1]: 2 lanes per row.
    int r = lane >> 1, part = lane & 1;
    float s = 0.f;
#pragma unroll 4
    for (int k = 0; k < 64; k++) s += hid[r * D2 + part * 64 + k] * W2[part * 64 + k];
    s += __shfl_xor(s, 1, 32);
    if (part == 0) {
        int row = row0 + r;
        if (row < nb_rows) logit[row] = s + b2[0];
    }
}

// ---------------------------------------------------------------------------
// 10) Scalar losses.
// ---------------------------------------------------------------------------
__global__ void k_scalars(const float* __restrict__ posb, const float* __restrict__ negb,
                          const float* __restrict__ cmt_acc,
                          float* __restrict__ out_cmt, float* __restrict__ out_reg) {
    __shared__ float red[256];
    float s = 0.f;
    for (int b = threadIdx.x; b < BB; b += 256) {
        float p = posb[b] + 1e-8f, q = negb[b] + 1e-8f;
        s += fabsf(p / (p + q) - 0.5f);
    }
    red[threadIdx.x] = s;
    __syncthreads();
    for (int o = 128; o > 0; o >>= 1) {
        if (threadIdx.x < (unsigned)o) red[threadIdx.x] += red[threadIdx.x + o];
        __syncthreads();
    }
    if (threadIdx.x == 0) {
        *out_reg = red[0] / (float)BB;
        *out_cmt = 0.1f * (*cmt_acc) / (float)(NN * DD);
    }
}

// ---------------------------------------------------------------------------
// Launcher
// ---------------------------------------------------------------------------
extern "C" void kernel_launch(void* const* d_in, const int* in_sizes, int n_in,
                              void* d_out, int out_size, void* d_ws, size_t ws_size,
                              hipStream_t stream) {
    const float* x       = (const float*)d_in[0];
    const int*   eidx    = (const int*)d_in[1];
    const int*   batch   = (const int*)d_in[2];
    const float* rW1 = (const float*)d_in[3],  *rb1 = (const float*)d_in[4];
    const float* rW2 = (const float*)d_in[5],  *rb2 = (const float*)d_in[6];
    const float* rg  = (const float*)d_in[7],  *rbb = (const float*)d_in[8];
    const float* reps = (const float*)d_in[9];
    const float* eW1 = (const float*)d_in[10], *eb1 = (const float*)d_in[11];
    const float* eW2 = (const float*)d_in[12], *eb2 = (const float*)d_in[13];
    const float* eg  = (const float*)d_in[14], *ebb = (const float*)d_in[15];
    const float* eeps = (const float*)d_in[16];
    const float* sW1 = (const float*)d_in[17], *sb1 = (const float*)d_in[18];
    const float* sg  = (const float*)d_in[19], *sb  = (const float*)d_in[20];
    const float* sW2 = (const float*)d_in[21], *sb2 = (const float*)d_in[22];
    const float* cW1 = (const float*)d_in[23], *cb1 = (const float*)d_in[24];
    const float* cg  = (const float*)d_in[25], *cbp = (const float*)d_in[26];
    const float* cW2 = (const float*)d_in[27], *cb2 = (const float*)d_in[28];
    const float* codebook = (const float*)d_in[29];

    const size_t ND = (size_t)NN * DD;
    float* h0    = (float*)d_ws;
    float* h1    = h0 + ND;
    float* agg   = h1 + ND;
    float* score = agg + ND;
    float* cbn   = score + ND;
    float* minT  = cbn + KK;
    int*   vqidx = (int*)(minT + NN);
    float* posb  = (float*)(vqidx + NN);
    float* negb  = posb + BB;
    float* cnt   = negb + BB;
    float* cpool = cnt + BB;
    float* spool = cpool + (size_t)BB * DD;
    float* cmt   = spool + (size_t)BB * DD;

    float* out = (float*)d_out;
    float* out_logit = out;                          // [B]
    float* out_cpool = out + BB;                     // [B*D]
    float* out_spool = out + BB + (size_t)BB * DD;   // [B*D]
    float* out_cmt   = out + BB + 2 * (size_t)BB * DD;
    float* out_reg   = out_cmt + 1;

    // Zero all accumulators (posb..cmt are contiguous).
    hipMemsetAsync(posb, 0, (3 * BB + 2 * (size_t)BB * DD + 1) * sizeof(float), stream);

    k_cb_norm<<<KK / 256, 256, 0, stream>>>(codebook, cbn);

    const int sc_blocks = (int)(((long long)EE * DD + 255) / 256);
    const int gin_blocks = NN / 16;   // N divisible by 16

    auto run_gnn = [&](const float* W1, const float* b1, const float* W2, const float* b2,
                       const float* g, const float* bb, const float* eps) {
        const float* hin = x;
        float* outs[3] = {h0, h1, h0};
        for (int l = 0; l < LLAY; l++) {
            hipMemsetAsync(agg, 0, ND * sizeof(float), stream);
            k_scatter_add<<<sc_blocks, 256, 0, stream>>>(hin, eidx, eidx + EE, agg, EE);
            k_gin_layer<<<gin_blocks, 32, 0, stream>>>(hin, agg, W1, b1, W2, b2, g, bb,
                                                       eps, l, (l < LLAY - 1) ? 1 : 0,
                                                       outs[l], NN);
            hin = outs[l];
        }
    };

    // Separator GNN -> nf_r in h0 -> score.
    run_gnn(rW1, rb1, rW2, rb2, rg, rbb, reps);
    k_sep_head<<<gin_blocks, 32, 0, stream>>>(h0, sW1, sb1, sg, sb, sW2, sb2, score, NN);
    k_score_reduce<<<(NN + 255) / 256, 256, 0, stream>>>(score, batch, posb, negb, cnt, NN);

    // Encoder GNN -> nf in h0.
    run_gnn(eW1, eb1, eW2, eb2, eg, ebb, eeps);

    // VQ + node finalize + pooling.
    k_vq<<<gin_blocks, 32, 0, stream>>>(h0, codebook, cbn, vqidx, minT, NN);
    k_finalize<<<(NN + 255) / 256, 256, 0, stream>>>(h0, codebook, vqidx, minT, score,
                                                     batch, cpool, spool, cmt, NN);
    k_pool_norm<<<(BB * DD + 255) / 256, 256, 0, stream>>>(cpool, spool, cnt,
                                                           out_cpool, out_spool, BB * DD);

    // Classifier head + scalar losses.
    k_classifier<<<BB / 16, 32, 0, stream>>>(out_cpool, cW1, cb1, cg, cbp, cW2, cb2,
                                             out_logit, BB);
    k_scalars<<<1, 256, 0, stream>>>(posb, negb, cmt, out_cmt, out_reg);
}